// GCN_78469052498323
// MI455X (gfx1250) — compile-verified
//
#include <hip/hip_runtime.h>

typedef __attribute__((ext_vector_type(2))) float v2f;
typedef __attribute__((ext_vector_type(8))) float v8f;

#define POOL_GRAPHS 256

// ---------------------------------------------------------------- utilities
__global__ void zero_f32(float* __restrict__ p, size_t n) {
  size_t i = (size_t)blockIdx.x * blockDim.x + threadIdx.x;
  if (i < n) p[i] = 0.0f;
}

// one float-add per endpoint per edge; deg buffers pre-zeroed
__global__ void degrees_kernel(const int* __restrict__ src, const int* __restrict__ dst,
                               float* __restrict__ degOut, float* __restrict__ degIn, int E) {
  int e = blockIdx.x * blockDim.x + threadIdx.x;
  if (e < E) {
    atomicAdd(&degOut[src[e]], 1.0f);
    atomicAdd(&degIn[dst[e]], 1.0f);
  }
}

// deg -> deg^{-1/2}, zero-degree -> 0 (matches reference)
__global__ void norm_inplace(float* __restrict__ p, size_t n) {
  size_t i = (size_t)blockIdx.x * blockDim.x + threadIdx.x;
  if (i < n) {
    float d = p[i];
    p[i] = (d > 0.0f) ? rsqrtf(d) : 0.0f;
  }
}

// ---------------------------------------------------------- SpMM scatter-add
// m[dst] += x[src] * norm_src[src]; one thread per (edge, 4-float chunk).
// Accumulator (<=51MB) is L2-resident (192MB L2), so atomics stay on-chip.
template <int D>
__global__ void aggregate_kernel(const float* __restrict__ x,
                                 const float* __restrict__ normSrc,
                                 const int* __restrict__ src,
                                 const int* __restrict__ dst,
                                 float* __restrict__ m, int E) {
  constexpr int CH = D / 4;
  size_t tid = (size_t)blockIdx.x * blockDim.x + threadIdx.x;
  size_t total = (size_t)E * CH;
  if (tid >= total) return;
  int e  = (int)(tid / CH);
  int c4 = (int)(tid % CH) * 4;
  int s = src[e];
  int d = dst[e];
  float ns = normSrc[s];
  const float4 v = *(const float4*)(x + (size_t)s * D + c4);
  float* o = m + (size_t)d * D + c4;
  atomicAdd(o + 0, v.x * ns);
  atomicAdd(o + 1, v.y * ns);
  atomicAdd(o + 2, v.z * ns);
  atomicAdd(o + 3, v.w * ns);
}

// ------------------------------------------------- WMMA GEMM + bias + ReLU
// out[16 rows x DOUT] = relu((m * norm_dst) @ W + b) using V_WMMA_F32_16X16X4_F32.
// One block per 16-node row tile; wave w owns output columns [16w, 16w+16).
// A tile staged in LDS with +1 padding (64-bank LDS -> conflict-free).
template <int DIN, int DOUT>
__global__ void gemm_relu_kernel(const float* __restrict__ m,
                                 const float* __restrict__ normDst,
                                 const float* __restrict__ W,
                                 const float* __restrict__ b,
                                 float* __restrict__ out) {
  constexpr int LDA = DIN + 1;
  __shared__ float tileA[16 * LDA];

  const int rowBase  = blockIdx.x * 16;
  const int tid      = threadIdx.x;
  const int nThreads = 32 * (DOUT / 16);

  // cooperative, coalesced load of the contiguous 16xDIN tile, norm_dst fused
  for (int i = tid; i < 16 * DIN; i += nThreads) {
    int r = i / DIN;
    int c = i % DIN;
    tileA[r * LDA + c] = m[(size_t)(rowBase + r) * DIN + c] * normDst[rowBase + r];
  }
  __syncthreads();

  const int wave    = tid >> 5;
  const int lane    = tid & 31;
  const int lh      = lane >> 4;   // 0: lanes 0-15, 1: lanes 16-31
  const int ll      = lane & 15;
  const int colBase = wave * 16;

  v8f acc = {};
#pragma unroll 4
  for (int k = 0; k < DIN; k += 4) {
    const int ka = k + lh * 2;     // ISA A layout: lo lanes K={k,k+1}, hi lanes K={k+2,k+3}
    v2f a, bv;
    a.x  = tileA[ll * LDA + ka];
    a.y  = tileA[ll * LDA + ka + 1];
    bv.x = W[(size_t)ka * DOUT + colBase + ll];        // B 4x16: lane = N column
    bv.y = W[(size_t)(ka + 1) * DOUT + colBase + ll];
    acc = __builtin_amdgcn_wmma_f32_16x16x4_f32(
        /*neg_a=*/false, a, /*neg_b=*/false, bv,
        /*c_mod=*/(short)0, acc, /*reuse_a=*/false, /*reuse_b=*/false);
  }

  const float bias = b[colBase + ll];
#pragma unroll
  for (int j = 0; j < 8; ++j) {    // C/D layout: VGPR j -> row j (+8 for hi lanes)
    const int row = rowBase + j + lh * 8;
    float v = acc[j] + bias;
    out[(size_t)row * DOUT + colBase + ll] = v > 0.0f ? v : 0.0f;
  }
}

// ------------------------------------------------------------ sum pooling
__global__ void pool_kernel(const float* __restrict__ h, const int* __restrict__ gid,
                            float* __restrict__ out, int n) {
  size_t tid = (size_t)blockIdx.x * blockDim.x + threadIdx.x;
  if (tid >= (size_t)n * 64) return;
  int v = (int)(tid >> 6);
  int f = (int)(tid & 63);
  atomicAdd(&out[(size_t)gid[v] * 64 + f], h[(size_t)v * 64 + f]);
}

// ---------------------------------------------------------------- launcher
extern "C" void kernel_launch(void* const* d_in, const int* in_sizes, int n_in,
                              void* d_out, int out_size, void* d_ws, size_t ws_size,
                              hipStream_t stream) {
  const float* h  = (const float*)d_in[0];
  const int* src  = (const int*)d_in[1];
  const int* dst  = (const int*)d_in[2];
  const int* gid  = (const int*)d_in[3];
  const float* W1 = (const float*)d_in[4];
  const float* b1 = (const float*)d_in[5];
  const float* W2 = (const float*)d_in[6];
  const float* b2 = (const float*)d_in[7];
  const float* W3 = (const float*)d_in[8];
  const float* b3 = (const float*)d_in[9];
  float* out = (float*)d_out;

  const int N = in_sizes[0] / 64;   // 100000
  const int E = in_sizes[1];        // 1600000

  // workspace layout
  float* normSrc = (float*)d_ws;              // N
  float* normDst = normSrc + N;               // N
  float* bufA    = normDst + N;               // N*128 (SpMM accumulator)
  float* bufB    = bufA + (size_t)N * 128;    // N*128 (GEMM output / next input)

  auto cdiv = [](size_t a, size_t b) { return (a + b - 1) / b; };
  const int T = 256;

  // degrees -> norms (norm buffers are contiguous: one zero + one rsqrt pass)
  zero_f32<<<(int)cdiv(2 * (size_t)N, T), T, 0, stream>>>(normSrc, 2 * (size_t)N);
  degrees_kernel<<<(int)cdiv(E, T), T, 0, stream>>>(src, dst, normSrc, normDst, E);
  norm_inplace<<<(int)cdiv(2 * (size_t)N, T), T, 0, stream>>>(normSrc, 2 * (size_t)N);

  const int rowTiles = N / 16;      // 6250, exact

  // ---- layer 1: d=64 -> 128
  zero_f32<<<(int)cdiv((size_t)N * 64, T), T, 0, stream>>>(bufA, (size_t)N * 64);
  aggregate_kernel<64><<<(int)cdiv((size_t)E * 16, T), T, 0, stream>>>(h, normSrc, src, dst, bufA, E);
  gemm_relu_kernel<64, 128><<<rowTiles, 256, 0, stream>>>(bufA, normDst, W1, b1, bufB);

  // ---- layer 2: d=128 -> 128
  zero_f32<<<(int)cdiv((size_t)N * 128, T), T, 0, stream>>>(bufA, (size_t)N * 128);
  aggregate_kernel<128><<<(int)cdiv((size_t)E * 32, T), T, 0, stream>>>(bufB, normSrc, src, dst, bufA, E);
  gemm_relu_kernel<128, 128><<<rowTiles, 256, 0, stream>>>(bufA, normDst, W2, b2, bufB);

  // ---- layer 3: d=128 -> 64
  zero_f32<<<(int)cdiv((size_t)N * 128, T), T, 0, stream>>>(bufA, (size_t)N * 128);
  aggregate_kernel<128><<<(int)cdiv((size_t)E * 32, T), T, 0, stream>>>(bufB, normSrc, src, dst, bufA, E);
  gemm_relu_kernel<128, 64><<<rowTiles, 128, 0, stream>>>(bufA, normDst, W3, b3, bufB);

  // ---- sum-pool readout
  zero_f32<<<(int)cdiv((size_t)POOL_GRAPHS * 64, T), T, 0, stream>>>(out, (size_t)POOL_GRAPHS * 64);
  pool_kernel<<<(int)cdiv((size_t)N * 64, T), T, 0, stream>>>(bufB, gid, out, N);
}